// GIN_64647847740123
// MI455X (gfx1250) — compile-verified
//
#include <hip/hip_runtime.h>

// ---------------------------------------------------------------------------
// GIN forward on MI455X (gfx1250, wave32).
//  - Feature map (12.8MB) is L2-resident (192MB L2): scatter/gather stay in L2.
//  - Aggregation: hardware fp32 atomics, one wave32 per edge, wave-uniform
//    index loads (scalarized src/dst).
//  - MLPs: V_WMMA_F32_16X16X4_F32 (fp32 matrix pipe). Weights are stored
//    K-transposed in LDS with rows padded to 68 floats so every WMMA fragment
//    is one conflict-free, 8B-aligned ds_load_b64 straight into an even VGPR
//    pair (no v_mov shuffles).
//  - Fused layer kernel dual-writes its output to the next layer's agg buffer,
//    eliminating the per-layer copy pass.
// ---------------------------------------------------------------------------

#define NNODES 50000
#define NEDGES 800000
#define DIM    64
#define MTILE  16
#define WPB    4                      // waves (16-row tiles) per block
#define NTILES (NNODES / MTILE)       // 3125 (exact)
#define PAD    68                     // LDS row stride (floats): 16B-aligned,
                                      // 4*col bank spread -> conflict-free b64

typedef __attribute__((ext_vector_type(2))) float v2f;
typedef __attribute__((ext_vector_type(8))) float v8f;

__device__ __forceinline__ v8f wmma_f32(v2f a, v2f b, v8f c) {
    // 8 args: (neg_a, A, neg_b, B, c_mod, C, reuse_a, reuse_b)
    return __builtin_amdgcn_wmma_f32_16x16x4_f32(
        false, a, false, b, (short)0, c, false, false);
}

// agg = x  (seed the "(1+eps)*x" self term for layer 0 only)
__global__ void gin_copy_kernel(const float4* __restrict__ src,
                                float4* __restrict__ dst) {
    int i = blockIdx.x * 256 + threadIdx.x;   // grid sized exactly: NNODES*DIM/4
    dst[i] = src[i];
}

// agg[dst] += h[src]; one wave32 per edge, lane l -> features 2l, 2l+1.
// Edge id is wave-uniform so src/dst loads scalarize to s_load.
__global__ void gin_scatter_kernel(const float* __restrict__ h,
                                   const int* __restrict__ edge,  // [2][NEDGES]
                                   float* __restrict__ agg) {
    const int e  = blockIdx.x * 8 + (threadIdx.x >> 5);   // uniform per wave
    const int d2 = (threadIdx.x & 31) * 2;
    const int s  = edge[e];
    const int t  = edge[NEDGES + e];
    const float2 v = *(const float2*)(h + s * DIM + d2);
    unsafeAtomicAdd(&agg[t * DIM + d2],     v.x);
    unsafeAtomicAdd(&agg[t * DIM + d2 + 1], v.y);
}

// FUSED=true : Out = relu( relu(BN(H@W1 + b1)) @ W2 + b2 ), optionally also
//              written to OutAgg (seed for next layer's aggregation).
// FUSED=false: Out = H@W1 + b1   (final Wout GEMM)
template <bool FUSED>
__global__ __launch_bounds__(128)
void gin_gemm_kernel(const float* __restrict__ H,
                     const float* __restrict__ W1, const float* __restrict__ b1,
                     const float* __restrict__ gamma, const float* __restrict__ beta,
                     const float* __restrict__ mu,    const float* __restrict__ var,
                     const float* __restrict__ W2, const float* __restrict__ b2,
                     float* __restrict__ Out, float* __restrict__ OutAgg) {
    __shared__ float sW1t[DIM * PAD];           // sW1t[n*PAD + k] = W1[k][n]
    __shared__ float sW2t[DIM * PAD];
    __shared__ float sP[4 * DIM];               // b1 | bn_scale | bn_shift | b2
    __shared__ float sH[WPB][MTILE * PAD];

    const int t = threadIdx.x;
    for (int i = t; i < DIM * DIM; i += 128) {  // K-transpose into LDS
        const int kk = i >> 6, n = i & 63;
        sW1t[n * PAD + kk] = W1[i];
        if constexpr (FUSED) sW2t[n * PAD + kk] = W2[i];
    }
    if (t < DIM) {
        sP[t] = b1[t];
        if constexpr (FUSED) {
            float sc = gamma[t] * rsqrtf(var[t] + 1e-5f);
            sP[DIM + t]     = sc;
            sP[2 * DIM + t] = beta[t] - mu[t] * sc;
            sP[3 * DIM + t] = b2[t];
        }
    }

    const int wave = t >> 5;
    const int lane = t & 31;
    const int tile = blockIdx.x * WPB + wave;   // wave-uniform -> EXEC stays full

    if (tile < NTILES) {                        // stage 16x64 input tile to LDS
        const float4* g4 = (const float4*)(H + (size_t)tile * MTILE * DIM);
        #pragma unroll
        for (int i = 0; i < 8; ++i) {
            const int idx = lane + 32 * i;      // 0..255 float4s
            const int row = idx >> 4, c4 = idx & 15;
            *(float4*)&sH[wave][row * PAD + c4 * 4] = g4[idx];
        }
    }
    __syncthreads();
    if (tile >= NTILES) return;

    const int half = lane >> 4;                 // lane half selects K+2 / M+8
    const int col  = lane & 15;
    const float* __restrict__ hT = &sH[wave][0];

    // ---- GEMM1: (16x64) @ (64x64), N split into 4 16-wide tiles -----------
    v8f acc[4] = {{}, {}, {}, {}};
    #pragma unroll
    for (int k = 0; k < DIM; k += 4) {
        const v2f a = *(const v2f*)(hT + col * PAD + k + 2 * half);
        #pragma unroll
        for (int nt = 0; nt < 4; ++nt) {
            const v2f b = *(const v2f*)(&sW1t[(nt * 16 + col) * PAD + k + 2 * half]);
            acc[nt] = wmma_f32(a, b, acc[nt]);
        }
    }

    if constexpr (FUSED) {
        // bias + BN(eval) + ReLU in D-layout, re-stripe through LDS to A-layout
        #pragma unroll
        for (int nt = 0; nt < 4; ++nt) {
            const int n = nt * 16 + col;
            const float bb = sP[n], sc = sP[DIM + n], sh = sP[2 * DIM + n];
            #pragma unroll
            for (int r = 0; r < 8; ++r) {
                float v = (acc[nt][r] + bb) * sc + sh;
                sH[wave][(r + 8 * half) * PAD + n] = fmaxf(v, 0.0f);
            }
        }
        // ---- GEMM2 -------------------------------------------------------
        v8f acc2[4] = {{}, {}, {}, {}};
        #pragma unroll
        for (int k = 0; k < DIM; k += 4) {
            const v2f a = *(const v2f*)(hT + col * PAD + k + 2 * half);
            #pragma unroll
            for (int nt = 0; nt < 4; ++nt) {
                const v2f b = *(const v2f*)(&sW2t[(nt * 16 + col) * PAD + k + 2 * half]);
                acc2[nt] = wmma_f32(a, b, acc2[nt]);
            }
        }
        #pragma unroll
        for (int nt = 0; nt < 4; ++nt) {
            const int n = nt * 16 + col;
            const float bb = sP[3 * DIM + n];
            #pragma unroll
            for (int r = 0; r < 8; ++r) {
                const float v = fmaxf(acc2[nt][r] + bb, 0.0f);  // outer ReLU
                const size_t o = (size_t)(tile * MTILE + r + 8 * half) * DIM + n;
                Out[o] = v;
                if (OutAgg) OutAgg[o] = v;      // seed next layer's agg
            }
        }
    } else {
        #pragma unroll
        for (int nt = 0; nt < 4; ++nt) {
            const int n = nt * 16 + col;
            const float bb = sP[n];
            #pragma unroll
            for (int r = 0; r < 8; ++r)
                Out[(size_t)(tile * MTILE + r + 8 * half) * DIM + n] =
                    acc[nt][r] + bb;
        }
    }
}

extern "C" void kernel_launch(void* const* d_in, const int* in_sizes, int n_in,
                              void* d_out, int out_size, void* d_ws, size_t ws_size,
                              hipStream_t stream) {
    const float* x    = (const float*)d_in[0];
    const int*   ei   = (const int*)  d_in[1];   // [2][800000] int32
    const float* W1s  = (const float*)d_in[2];   // [4][64][64]
    const float* b1s  = (const float*)d_in[3];
    const float* gms  = (const float*)d_in[4];
    const float* bts  = (const float*)d_in[5];
    const float* mus  = (const float*)d_in[6];
    const float* vrs  = (const float*)d_in[7];
    const float* W2s  = (const float*)d_in[8];
    const float* b2s  = (const float*)d_in[9];
    const float* Wout = (const float*)d_in[10];
    const float* bout = (const float*)d_in[11];

    float* agg  = (float*)d_ws;                  // 12.8 MB scatter accumulator
    float* hbuf = agg + (size_t)NNODES * DIM;    // 12.8 MB ping buffer

    const int copyBlocks = (NNODES * DIM / 4) / 256;       // 3125, exact
    const int scatBlocks = NEDGES / 8;                     // 100000, exact
    const int gemmBlocks = (NTILES + WPB - 1) / WPB;       // 782

    const float* hin = x;
    for (int l = 0; l < 4; ++l) {
        if (l == 0)                              // seed agg = x (layer 0 only;
            gin_copy_kernel<<<copyBlocks, 256, 0, stream>>>(   // later layers
                (const float4*)hin, (float4*)agg);             // dual-write)
        gin_scatter_kernel<<<scatBlocks, 256, 0, stream>>>(hin, ei, agg);
        gin_gemm_kernel<true><<<gemmBlocks, 128, 0, stream>>>(
            agg, W1s + l * DIM * DIM, b1s + l * DIM,
            gms + l * DIM, bts + l * DIM, mus + l * DIM, vrs + l * DIM,
            W2s + l * DIM * DIM, b2s + l * DIM, hbuf,
            (l < 3) ? agg : nullptr);            // agg seed for next layer
        hin = hbuf;                              // hbuf consumed before rewrite
    }
    gin_gemm_kernel<false><<<gemmBlocks, 128, 0, stream>>>(
        hin, Wout, bout, nullptr, nullptr, nullptr, nullptr, nullptr, nullptr,
        (float*)d_out, nullptr);
}